// Voxelization_39505109188919
// MI455X (gfx1250) — compile-verified
//
#include <hip/hip_runtime.h>
#include <hip/hip_bf16.h>
#include <stdint.h>

// ---------------------------------------------------------------------------
// Hard voxelization for MI455X (gfx1250, wave32).
//
// Grid 1408x1600x40 (90.1M cells), ~3.4M distinct occupied voxels from 4M
// points (mostly singletons). L2-resident AoS hash: 2^22 entries x 32B
// (slots0-3 | key | slot4 | evid | pad) = 134MB < 192MB L2, so every
// CAS/atomicMin resolves in L2 and each voxel costs ~1 cacheline per pass.
// Deterministic top-5 atomicMin insertion cascade reproduces the reference's
// stable-sort slot assignment; FCFS voxel ids come from a prefix sum of
// "creator" marks in original point order.
// ---------------------------------------------------------------------------

#define GX 1408
#define GY 1600
#define GZ 40
#define MAXP 5
#define HC_LOG 22
#define HC (1u << HC_LOG)
#define HSHIFT (32u - HC_LOG)
#define EMPTY 0xFFFFFFFFu
#define PPB 4096          // points per block in scan passes (256 thr * 16)

// Entry dword offsets within a 8-dword (32B) hash entry.
#define E_S0   0          // slots 0..3 at [0..3] (16B aligned -> uint4 load)
#define E_KEY  4
#define E_S4   5
#define E_VID  6

__device__ __forceinline__ int slot_off(int s) { return (s < 4) ? s : E_S4; }

__device__ __forceinline__ bool compute_cell(float x, float y, float z,
                                             int& cx, int& cy, int& cz) {
    // Must match jnp.floor((p - lo) / voxel_size) in f32 exactly:
    // IEEE f32 subtract, IEEE f32 divide, floorf.
    float qx = (x - 0.0f)    / 0.05f;
    float qy = (y - (-40.0f))/ 0.05f;
    float qz = (z - (-3.0f)) / 0.1f;
    cx = (int)floorf(qx);
    cy = (int)floorf(qy);
    cz = (int)floorf(qz);
    return (cx >= 0) & (cx < GX) & (cy >= 0) & (cy < GY) & (cz >= 0) & (cz < GZ);
}

// ---------------------------------------------------------------------------
// Pass 1: hash-insert each valid point's voxel; deterministic top-5 cascade
// of smallest original point indices (lock-free insertion sort via atomicMin).
// pent[i] caches the entry index so later passes never re-probe.
// ---------------------------------------------------------------------------
__global__ void k_build(const float4* __restrict__ pts, int N,
                        uint32_t* __restrict__ tab,
                        uint32_t* __restrict__ pent) {
    int i = blockIdx.x * blockDim.x + threadIdx.x;
    if (i >= N) return;
    // gfx1250 prefetch path (global_prefetch_b8): pull next tile toward L2.
    int pf = i + 16384; if (pf >= N) pf = i;
    __builtin_prefetch(&pts[pf], 0, 1);

    float4 p = pts[i];
    int cx, cy, cz;
    if (!compute_cell(p.x, p.y, p.z, cx, cy, cz)) { pent[i] = EMPTY; return; }
    uint32_t lin = (uint32_t)cz * (GY * GX) + (uint32_t)cy * GX + (uint32_t)cx;

    // Open addressing, triangular probing (full-cycle on pow2 tables).
    uint32_t h    = (lin * 2654435761u) >> HSHIFT;
    uint32_t step = 1u;
    uint32_t e    = h * 8u;
    for (;;) {
        uint32_t k = atomicCAS(&tab[e + E_KEY], EMPTY, lin);
        if (k == EMPTY || k == lin) break;
        h = (h + step) & (HC - 1u);
        step++;
        e = h * 8u;
    }
    pent[i] = h;

    // Deterministic top-5 smallest-index cascade.
    uint32_t cur = (uint32_t)i;
    #pragma unroll
    for (int s = 0; s < MAXP; s++) {
        uint32_t old = atomicMin(&tab[e + slot_off(s)], cur);
        if (old == EMPTY) break;        // filled a fresh slot, done
        if (old > cur) cur = old;       // we landed; displaced value sinks
        // else cur keeps probing deeper slots
    }
}

// ---------------------------------------------------------------------------
// Helper: vectorized read of 16 pent entries per thread.
// ---------------------------------------------------------------------------
__device__ __forceinline__ void load_pent16(const uint32_t* __restrict__ pent,
                                            int base, int N, uint32_t hv[16]) {
    if (base + 16 <= N) {
        const uint4* p4 = (const uint4*)(pent + base);  // 16B aligned
        #pragma unroll
        for (int q = 0; q < 4; q++) {
            uint4 v = p4[q];
            hv[4*q+0] = v.x; hv[4*q+1] = v.y; hv[4*q+2] = v.z; hv[4*q+3] = v.w;
        }
    } else {
        #pragma unroll
        for (int j = 0; j < 16; j++)
            hv[j] = (base + j < N) ? pent[base + j] : EMPTY;
    }
}

// ---------------------------------------------------------------------------
// Pass 2: per-block creator counts (creator <=> slot0 of entry == point idx).
// ---------------------------------------------------------------------------
__global__ void k_marksum(const uint32_t* __restrict__ pent,
                          const uint32_t* __restrict__ tab,
                          int N, uint32_t* __restrict__ bsum) {
    __shared__ uint32_t sh[256];
    int t = threadIdx.x;
    int base = blockIdx.x * PPB + t * 16;
    uint32_t hv[16];
    load_pent16(pent, base, N, hv);
    uint32_t s = 0;
    #pragma unroll
    for (int j = 0; j < 16; j++) {
        uint32_t h = hv[j];
        if (h != EMPTY && tab[h * 8u + E_S0] == (uint32_t)(base + j)) s++;
    }
    sh[t] = s;
    __syncthreads();
    for (int off = 128; off > 0; off >>= 1) {
        if (t < off) sh[t] += sh[t + off];
        __syncthreads();
    }
    if (t == 0) bsum[blockIdx.x] = sh[0];
}

// ---------------------------------------------------------------------------
// Pass 3: single-block exclusive scan of block sums; writes voxel_num.
// ---------------------------------------------------------------------------
__global__ void k_scan(const uint32_t* __restrict__ bsum,
                       uint32_t* __restrict__ boff, int NB,
                       const int* __restrict__ maxVoxPtr,
                       float* __restrict__ out_vnum) {
    __shared__ uint32_t sh[1024];
    int t = threadIdx.x;
    int CH = (NB + 1023) / 1024;
    uint32_t s = 0;
    for (int j = 0; j < CH; j++) {
        int idx = t * CH + j;
        if (idx < NB) s += bsum[idx];
    }
    sh[t] = s;
    __syncthreads();
    for (int off = 1; off < 1024; off <<= 1) {
        uint32_t v = (t >= off) ? sh[t - off] : 0u;
        __syncthreads();
        sh[t] += v;
        __syncthreads();
    }
    uint32_t total = sh[1023];
    uint32_t run = (t == 0) ? 0u : sh[t - 1];
    for (int j = 0; j < CH; j++) {
        int idx = t * CH + j;
        if (idx < NB) { boff[idx] = run; run += bsum[idx]; }
    }
    if (t == 0) {
        uint32_t mv = (uint32_t)(*maxVoxPtr);
        *out_vnum = (float)(total < mv ? total : mv);
    }
}

// ---------------------------------------------------------------------------
// Pass 4: FCFS voxel id = exclusive rank of creator among creators in
// original point order. Creator writes vid into its entry, coors (xyz) and
// num_points (= #filled slots = min(count, MAXP)).
// ---------------------------------------------------------------------------
__global__ void k_assign(const float4* __restrict__ pts,
                         const uint32_t* __restrict__ pent,
                         uint32_t* __restrict__ tab,
                         const uint32_t* __restrict__ boff, int N,
                         const int* __restrict__ maxVoxPtr,
                         float* __restrict__ out_coors,
                         float* __restrict__ out_npts) {
    __shared__ uint32_t sh[256];
    int t = threadIdx.x;
    int base = blockIdx.x * PPB + t * 16;
    uint32_t hv[16];
    load_pent16(pent, base, N, hv);
    uint32_t mbits = 0, cnt = 0;
    #pragma unroll
    for (int j = 0; j < 16; j++) {
        uint32_t h = hv[j];
        if (h != EMPTY && tab[h * 8u + E_S0] == (uint32_t)(base + j)) {
            mbits |= (1u << j);
            cnt++;
        }
    }
    sh[t] = cnt;
    __syncthreads();
    for (int off = 1; off < 256; off <<= 1) {
        uint32_t v = (t >= off) ? sh[t - off] : 0u;
        __syncthreads();
        sh[t] += v;
        __syncthreads();
    }
    uint32_t run = boff[blockIdx.x] + ((t == 0) ? 0u : sh[t - 1]);
    uint32_t mv = (uint32_t)(*maxVoxPtr);
    #pragma unroll
    for (int j = 0; j < 16; j++) {
        if (mbits & (1u << j)) {
            int i = base + j;
            uint32_t e = hv[j] * 8u;
            uint32_t v = run++;
            tab[e + E_VID] = v;
            if (v < mv) {
                float4 p = pts[i];
                int cx, cy, cz;
                compute_cell(p.x, p.y, p.z, cx, cy, cz);
                out_coors[(size_t)v * 3 + 0] = (float)cx;
                out_coors[(size_t)v * 3 + 1] = (float)cy;
                out_coors[(size_t)v * 3 + 2] = (float)cz;
                uint4 s03 = *(const uint4*)&tab[e + E_S0];   // 16B aligned
                uint32_t n = (s03.x != EMPTY) + (s03.y != EMPTY) +
                             (s03.z != EMPTY) + (s03.w != EMPTY) +
                             (tab[e + E_S4] != EMPTY);
                out_npts[v] = (float)n;
            }
        }
    }
}

// ---------------------------------------------------------------------------
// Pass 5: every point ranked in its voxel's top-5 scatters its 4 channels.
// Entry read = two global_load_b128 (slots0-3, then key|slot4|evid|pad).
// ---------------------------------------------------------------------------
__global__ void k_scatter(const float4* __restrict__ pts,
                          const uint32_t* __restrict__ pent,
                          const uint32_t* __restrict__ tab, int N,
                          const int* __restrict__ maxVoxPtr,
                          float* __restrict__ out_voxels) {
    int i = blockIdx.x * blockDim.x + threadIdx.x;
    if (i >= N) return;
    uint32_t h = pent[i];
    if (h == EMPTY) return;
    uint32_t e = h * 8u;
    uint4 s03 = *(const uint4*)&tab[e + E_S0];       // slots 0..3
    uint4 kse = *(const uint4*)&tab[e + E_KEY];      // key, slot4, evid, pad
    uint32_t me = (uint32_t)i;
    int r = -1;
    if      (s03.x == me) r = 0;
    else if (s03.y == me) r = 1;
    else if (s03.z == me) r = 2;
    else if (s03.w == me) r = 3;
    else if (kse.y == me) r = 4;
    if (r < 0) return;                               // not among 5 smallest
    uint32_t v = kse.z;                              // evid
    uint32_t mv = (uint32_t)(*maxVoxPtr);
    if (v >= mv) return;
    float4 p = pts[i];
    float* dst = out_voxels + ((size_t)v * MAXP + r) * 4;
    dst[0] = p.x; dst[1] = p.y; dst[2] = p.z; dst[3] = p.w;
}

// ---------------------------------------------------------------------------
extern "C" void kernel_launch(void* const* d_in, const int* in_sizes, int n_in,
                              void* d_out, int out_size, void* d_ws, size_t ws_size,
                              hipStream_t stream) {
    const float4*  pts    = (const float4*)d_in[0];
    const int*     maxPts = (const int*)d_in[1]; (void)maxPts;   // == MAXP
    const int*     maxVox = (const int*)d_in[2];
    float*         out    = (float*)d_out;

    const int N  = in_sizes[0] / 4;
    const int NB = (N + PPB - 1) / PPB;

    // Workspace layout: tab (HC entries x 32B) | pent (N) | bsum | boff.
    size_t need = (size_t)HC * 8 * 4 + (size_t)N * 4 + (size_t)NB * 8 + 256;
    if (need > ws_size) return;              // cannot run without scratch
    uint8_t* w = (uint8_t*)d_ws;
    uint32_t* tab  = (uint32_t*)w;  w += (size_t)HC * 8 * 4;
    uint32_t* pent = (uint32_t*)w;  w += (size_t)N * 4;
    uint32_t* bsum = (uint32_t*)w;  w += (size_t)NB * 4;
    uint32_t* boff = (uint32_t*)w;

    // Output layout (flat float): voxels | coors | num_points | voxel_num.
    // out_size = mv*MAXP*4 + mv*3 + mv + 1 = 24*mv + 1.
    int mv = (out_size - 1) / 24;
    float* out_voxels = out;
    float* out_coors  = out + (size_t)mv * MAXP * 4;
    float* out_npts   = out_coors + (size_t)mv * 3;
    float* out_vnum   = out_npts + mv;

    // Init: whole hash table (keys + slots; vid overwritten before use) to
    // 0xFFFFFFFF; outputs to zero padding. Both graph-capture safe.
    hipMemsetAsync(tab, 0xFF, (size_t)HC * 8 * 4, stream);
    hipMemsetAsync(d_out, 0, (size_t)out_size * sizeof(float), stream);

    k_build  <<<(N + 255) / 256, 256, 0, stream>>>(pts, N, tab, pent);
    k_marksum<<<NB, 256, 0, stream>>>(pent, tab, N, bsum);
    k_scan   <<<1, 1024, 0, stream>>>(bsum, boff, NB, maxVox, out_vnum);
    k_assign <<<NB, 256, 0, stream>>>(pts, pent, tab, boff, N,
                                      maxVox, out_coors, out_npts);
    k_scatter<<<(N + 255) / 256, 256, 0, stream>>>(pts, pent, tab, N,
                                                   maxVox, out_voxels);
}